// GCNReaonser_49855980372020
// MI455X (gfx1250) — compile-verified
//
#include <hip/hip_runtime.h>
#include <math.h>

#define NN 20000
#define EE 160000
#define BB 16
#define SS 32
#define DD 128
#define KK 4
#define LL 4
#define TT 3

typedef float v2f __attribute__((ext_vector_type(2)));
typedef float v8f __attribute__((ext_vector_type(8)));
typedef unsigned int u32x4 __attribute__((ext_vector_type(4)));
typedef int i32x4 __attribute__((ext_vector_type(4)));
typedef int i32x8 __attribute__((ext_vector_type(8)));

#if defined(__has_builtin)
#if __has_builtin(__builtin_amdgcn_tensor_load_to_lds) && \
    __has_builtin(__builtin_amdgcn_s_wait_tensorcnt)
#define HAVE_TDM 1
#endif
#endif
#ifndef HAVE_TDM
#define HAVE_TDM 0
#endif

// ---------------- generic helpers ----------------

__global__ void zero_kernel(float* __restrict__ p, int n) {
  int i = blockIdx.x * blockDim.x + threadIdx.x;
  int stride = gridDim.x * blockDim.x;
  for (; i < n; i += stride) p[i] = 0.0f;
}

__global__ void copy_kernel(float* __restrict__ dst, const float* __restrict__ src, int n) {
  int i = blockIdx.x * blockDim.x + threadIdx.x;
  int stride = gridDim.x * blockDim.x;
  for (; i < n; i += stride) dst[i] = src[i];
}

// ---------------- init_instructions ----------------
__global__ void init_ins_kernel(const float* __restrict__ query,
                                const float* __restrict__ query_mask,
                                const float* __restrict__ query_pool,
                                const float* __restrict__ W_v,
                                const float* __restrict__ W_u,
                                float* __restrict__ ins) {
  int b = blockIdx.x;
  int d = threadIdx.x;
  __shared__ float s_q[DD], s_ip[DD], s_cat[4 * DD], s_qk[DD], s_sc[SS], s_wu[DD];
  s_q[d] = query_pool[b * DD + d];
  s_ip[d] = 0.0f;
  s_wu[d] = W_u[d];
  __syncthreads();
  for (int k = 0; k < KK; ++k) {
    s_cat[d]          = s_ip[d];
    s_cat[DD + d]     = s_q[d];
    s_cat[2 * DD + d] = s_q[d] * s_ip[d];
    s_cat[3 * DD + d] = s_q[d] - s_ip[d];
    __syncthreads();
    float acc = 0.0f;
    const float* wv = W_v + (size_t)k * 4 * DD * DD;
    for (int f = 0; f < 4 * DD; ++f) acc += s_cat[f] * wv[f * DD + d];
    s_qk[d] = acc;
    __syncthreads();
    if (d < SS) {
      float sc = 0.0f;
      const float* qr = query + ((size_t)b * SS + d) * DD;
      for (int j = 0; j < DD; ++j) sc += s_qk[j] * qr[j] * s_wu[j];
      if (query_mask[b * SS + d] == 0.0f) sc = -1.0e9f;
      s_sc[d] = sc;
    }
    __syncthreads();
    if (d == 0) {
      float m = -1.0e30f;
      for (int s2 = 0; s2 < SS; ++s2) m = fmaxf(m, s_sc[s2]);
      float sum = 0.0f;
      for (int s2 = 0; s2 < SS; ++s2) { s_sc[s2] = __expf(s_sc[s2] - m); sum += s_sc[s2]; }
      float inv = 1.0f / sum;
      for (int s2 = 0; s2 < SS; ++s2) s_sc[s2] *= inv;
    }
    __syncthreads();
    float ik = 0.0f;
    for (int s2 = 0; s2 < SS; ++s2) {
      float mk = query_mask[b * SS + s2];
      ik += s_sc[s2] * query[((size_t)b * SS + s2) * DD + d] * mk;
    }
    ins[((size_t)b * KK + k) * DD + d] = ik;
    __syncthreads();
    s_ip[d] = ik;
    __syncthreads();
  }
}

// ---------------- Tensor Data Mover: 2D tile (global -> LDS) ----------------
// Builds the D# per CDNA5 ISA and issues TENSOR_LOAD_TO_LDS.
// tensor dims are set equal to the tile dims (tile fully in-bounds); the real
// matrix leading dimension goes into tensor_dim0_stride.
#if HAVE_TDM
__device__ __forceinline__ void tdm_load_2d(unsigned lds_addr, const void* gptr,
                                            unsigned tile_x, unsigned tile_y,
                                            unsigned long long stride_elems) {
  unsigned long long ga = (unsigned long long)gptr;
  u32x4 g0;
  g0[0] = 1u;                                    // count=1, user mode
  g0[1] = lds_addr;                              // LDS byte address
  g0[2] = (unsigned)ga;                          // global_addr[31:0]
  g0[3] = (unsigned)((ga >> 32) & 0x1FFFFFFu) | (2u << 30);  // addr[56:32] | type=2
  i32x8 g1;
  g1[0] = (int)(2u << 16);                       // data_size=2 (4 bytes), wg_mask=0
  g1[1] = (int)((tile_x & 0xFFFFu) << 16);       // tensor_dim0[15:0] (== tile_x)
  g1[2] = (int)((tile_x >> 16) | ((tile_y & 0xFFFFu) << 16));   // dim0 hi | dim1 lo
  g1[3] = (int)((tile_y >> 16) | (tile_x << 16));               // dim1 hi | tile_dim0
  g1[4] = (int)tile_y;                           // tile_dim1 | tile_dim2=0
  g1[5] = (int)(unsigned)(stride_elems & 0xFFFFFFFFull);        // dim0_stride lo
  g1[6] = (int)(unsigned)((stride_elems >> 32) & 0xFFFFull);    // dim0_stride hi | dim1_stride lo
  g1[7] = 0;
  i32x4 z4 = {0, 0, 0, 0};
#if defined(__clang_major__) && (__clang_major__ >= 23)
  i32x8 z8 = {0, 0, 0, 0, 0, 0, 0, 0};
  __builtin_amdgcn_tensor_load_to_lds(g0, g1, z4, z4, z8, 0);
#else
  __builtin_amdgcn_tensor_load_to_lds(g0, g1, z4, z4, 0);
#endif
}
#endif

// ---------------- WMMA f32 GEMM with LDS-staged panels ----------------
// C(M x DD) = A(M x KTOT) * B(KTOT x DD). One block per 16-row stripe of C;
// 8 waves cover the 8 column tiles. A/B panels (K-depth 32) are staged into
// LDS (TDM double-buffered when available) and consumed by
// V_WMMA_F32_16X16X4_F32 steps.
// SPLIT: A columns [0,DD) from A0 (ld=DD), [DD,KTOT) from A1 (ld=KK*DD).
template <int KTOT, bool RELU, bool SPLIT>
__global__ __launch_bounds__(256) void gemm_wmma_kernel(
    const float* __restrict__ A0, const float* __restrict__ A1,
    const float* __restrict__ Bw, float* __restrict__ C, int M) {
  constexpr int PANEL = 32;
  constexpr int P = KTOT / PANEL;
  __shared__ float sA[2][16 * PANEL];
  __shared__ float sB[2][PANEL * DD];

  const int wave = threadIdx.x >> 5;  // == column tile tn
  const int lane = threadIdx.x & 31;
  const int tm = blockIdx.x;
  if (tm * 16 >= M) return;  // never taken for M=20000 (uniform anyway)
  const int half = lane >> 4;
  const int lm = lane & 15;
  const int ncol = wave * 16 + lm;

  v8f c = {0.f, 0.f, 0.f, 0.f, 0.f, 0.f, 0.f, 0.f};

#if HAVE_TDM
  // wave 0 drives the Tensor Data Mover; double-buffered panels.
  if (threadIdx.x < 32) {
    const float* srcA = A0 + (size_t)tm * 16 * DD;
    tdm_load_2d((unsigned)(unsigned long long)&sA[0][0], srcA, PANEL, 16, DD);
    tdm_load_2d((unsigned)(unsigned long long)&sB[0][0], Bw, DD, PANEL, DD);
  }
  for (int p = 0; p < P; ++p) {
    const int buf = p & 1;
    if (threadIdx.x < 32) {
      if (p + 1 < P) {
        const int col0 = (p + 1) * PANEL;
        const float* srcA;
        unsigned long long ldA;
        if (!SPLIT || col0 < DD) {
          srcA = A0 + (size_t)tm * 16 * DD + col0;
          ldA = DD;
        } else {
          srcA = A1 + (size_t)tm * 16 * (KK * DD) + (col0 - DD);
          ldA = KK * DD;
        }
        tdm_load_2d((unsigned)(unsigned long long)&sA[buf ^ 1][0], srcA, PANEL, 16, ldA);
        tdm_load_2d((unsigned)(unsigned long long)&sB[buf ^ 1][0],
                    Bw + (size_t)col0 * DD, DD, PANEL, DD);
        __builtin_amdgcn_s_wait_tensorcnt(2);  // oldest 2 (panel p) complete
      } else {
        __builtin_amdgcn_s_wait_tensorcnt(0);
      }
    }
    __syncthreads();  // panel p visible to all waves
    const float* Al = &sA[buf][0];
    const float* Bl = &sB[buf][0];
#pragma unroll
    for (int ks = 0; ks < PANEL; ks += 4) {
      const int ka = ks + 2 * half;
      v2f a, b;
      a.x = Al[lm * PANEL + ka];
      a.y = Al[lm * PANEL + ka + 1];
      b.x = Bl[ka * DD + ncol];
      b.y = Bl[(ka + 1) * DD + ncol];
      c = __builtin_amdgcn_wmma_f32_16x16x4_f32(false, a, false, b, (short)0, c,
                                                false, false);
    }
    __syncthreads();  // all waves done with buf before TDM overwrites it
  }
#else
  // cooperative-load fallback (no TDM builtin available)
  for (int p = 0; p < P; ++p) {
    const int col0 = p * PANEL;
    const float* srcA;
    int ldA;
    if (!SPLIT || col0 < DD) {
      srcA = A0 + (size_t)tm * 16 * DD + col0;
      ldA = DD;
    } else {
      srcA = A1 + (size_t)tm * 16 * (KK * DD) + (col0 - DD);
      ldA = KK * DD;
    }
    for (int i = threadIdx.x; i < 16 * PANEL; i += 256)
      sA[0][i] = srcA[(i / PANEL) * ldA + (i % PANEL)];
    const float* srcB = Bw + (size_t)col0 * DD;
    for (int i = threadIdx.x; i < PANEL * DD; i += 256) sB[0][i] = srcB[i];
    __syncthreads();
#pragma unroll
    for (int ks = 0; ks < PANEL; ks += 4) {
      const int ka = ks + 2 * half;
      v2f a, b;
      a.x = sA[0][lm * PANEL + ka];
      a.y = sA[0][lm * PANEL + ka + 1];
      b.x = sB[0][ka * DD + ncol];
      b.y = sB[0][(ka + 1) * DD + ncol];
      c = __builtin_amdgcn_wmma_f32_16x16x4_f32(false, a, false, b, (short)0, c,
                                                false, false);
    }
    __syncthreads();
  }
#endif

#pragma unroll
  for (int v = 0; v < 8; ++v) {
    const int r = tm * 16 + v + 8 * half;
    float val = c[v];
    if (RELU) val = fmaxf(val, 0.0f);
    C[(size_t)r * DD + ncol] = val;
  }
}

// ---------------- edge scatter: agg[dst] += p[src]*relu(ins[batch[src]]*th[src]) ----------------
__global__ void scatter_kernel(const int* __restrict__ edge_index,
                               const int* __restrict__ batch_idx,
                               const float* __restrict__ ins,
                               const float* __restrict__ th,
                               const float* __restrict__ p,
                               float* __restrict__ agg) {
  int e = blockIdx.x;
  int s = edge_index[e];           // src row
  int dnode = edge_index[EE + e];  // dst row
  int b = batch_idx[s];
  float ps = p[s];
  const float* insb = ins + (size_t)b * KK * DD;
  const float* ths = th + (size_t)s * DD;
  float* ag = agg + (size_t)dnode * KK * DD;
  for (int idx = threadIdx.x; idx < KK * DD; idx += 256) {
    float v = insb[idx] * ths[idx & (DD - 1)];
    v = v > 0.0f ? v : 0.0f;
    atomicAdd(&ag[idx], v * ps);
  }
}

// ---------------- logits[i] = h_new[i] . w ----------------
__global__ void logits_kernel(const float* __restrict__ h,
                              const float* __restrict__ w,
                              float* __restrict__ logits, int n) {
  int wave = threadIdx.x >> 5;
  int lane = threadIdx.x & 31;
  int row = blockIdx.x * (blockDim.x >> 5) + wave;
  if (row >= n) return;
  const float* hr = h + (size_t)row * DD;
  float acc = 0.0f;
  for (int j = lane; j < DD; j += 32) acc += hr[j] * w[j];
  for (int off = 16; off > 0; off >>= 1) acc += __shfl_down(acc, off, 32);
  if (lane == 0) logits[row] = acc;
}

// ---------------- global softmax over N ----------------
__global__ void softmax_reduce_kernel(const float* __restrict__ logits,
                                      float* __restrict__ stats, int n) {
  __shared__ float sd[1024];
  int t = threadIdx.x;
  float m = -1.0e30f;
  for (int i = t; i < n; i += 1024) m = fmaxf(m, logits[i]);
  sd[t] = m;
  __syncthreads();
  for (int off = 512; off > 0; off >>= 1) {
    if (t < off) sd[t] = fmaxf(sd[t], sd[t + off]);
    __syncthreads();
  }
  float mx = sd[0];
  __syncthreads();
  float s = 0.0f;
  for (int i = t; i < n; i += 1024) s += __expf(logits[i] - mx);
  sd[t] = s;
  __syncthreads();
  for (int off = 512; off > 0; off >>= 1) {
    if (t < off) sd[t] += sd[t + off];
    __syncthreads();
  }
  if (t == 0) { stats[0] = mx; stats[1] = sd[0]; }
}

__global__ void softmax_norm_kernel(const float* __restrict__ logits,
                                    const float* __restrict__ stats,
                                    float* __restrict__ p, int n) {
  int i = blockIdx.x * blockDim.x + threadIdx.x;
  if (i < n) p[i] = __expf(logits[i] - stats[0]) / stats[1];
}

// ---------------- h_e[b] = sum_{i: batch=b, p0[i]>0} h[i] ----------------
__global__ void he_accum_kernel(const float* __restrict__ h,
                                const float* __restrict__ p0,
                                const int* __restrict__ batch_idx,
                                float* __restrict__ he) {
  int i = blockIdx.x;
  int d = threadIdx.x;
  if (p0[i] > 0.0f) {
    int b = batch_idx[i];
    atomicAdd(&he[b * DD + d], h[(size_t)i * DD + d]);
  }
}

// ---------------- instruction GRU-style update (in place) ----------------
__global__ void ins_update_kernel(float* __restrict__ ins,
                                  const float* __restrict__ he,
                                  const float* __restrict__ W_q,
                                  const float* __restrict__ gate) {
  int bk = blockIdx.x;  // B*K blocks
  int b = bk / KK, k = bk % KK;
  int d = threadIdx.x;
  __shared__ float s_cat[4 * DD];
  float iv = ins[(size_t)bk * DD + d];
  float hv = he[b * DD + d];
  s_cat[d]          = iv;
  s_cat[DD + d]     = hv;
  s_cat[2 * DD + d] = iv - hv;
  s_cat[3 * DD + d] = iv * hv;
  __syncthreads();
  float tr = 0.0f, gs = 0.0f;
  const float* gk = gate + (size_t)k * 4 * DD * DD;
  for (int f = 0; f < 4 * DD; ++f) {
    float cv = s_cat[f];
    tr += cv * W_q[f * DD + d];
    gs += cv * gk[f * DD + d];
  }
  float g = 1.0f / (1.0f + __expf(-gs));
  ins[(size_t)bk * DD + d] = (1.0f - g) * iv + g * tr;
}

// ---------------- host orchestration ----------------

extern "C" void kernel_launch(void* const* d_in, const int* in_sizes, int n_in,
                              void* d_out, int out_size, void* d_ws, size_t ws_size,
                              hipStream_t stream) {
  (void)in_sizes; (void)n_in; (void)out_size; (void)ws_size;
  const float* x          = (const float*)d_in[0];
  const float* query      = (const float*)d_in[1];
  const float* query_mask = (const float*)d_in[2];
  const float* query_pool = (const float*)d_in[3];
  const float* p_0        = (const float*)d_in[4];
  const int*   batch_idx  = (const int*)d_in[5];
  const int*   edge_index = (const int*)d_in[6];
  const float* W_v        = (const float*)d_in[7];
  const float* gate       = (const float*)d_in[8];
  const float* W_u        = (const float*)d_in[9];
  const float* W_q        = (const float*)d_in[10];
  const float* W_x        = (const float*)d_in[11];
  const float* W_h        = (const float*)d_in[12];
  const float* w          = (const float*)d_in[13];
  float* out = (float*)d_out;

  // workspace layout (floats); total ~18.0M floats (~72 MB)
  float* ws     = (float*)d_ws;
  float* ins    = ws;                          // B*K*D   = 8192
  float* he     = ins + BB * KK * DD;          // B*D     = 2048
  float* stats  = he + BB * DD;                // 16 (pad)
  float* logits = stats + 16;                  // N
  float* pbuf   = logits + NN;                 // N
  float* hA     = pbuf + NN;                   // N*D
  float* hB     = hA + (size_t)NN * DD;        // N*D
  float* th     = hB + (size_t)NN * DD;        // N*D
  float* agg    = th + (size_t)NN * DD;        // N*K*D

  init_ins_kernel<<<BB, DD, 0, stream>>>(query, query_mask, query_pool, W_v, W_u, ins);
  copy_kernel<<<512, 256, 0, stream>>>(hA, x, NN * DD);

  float* hcur = hA;
  float* hnxt = hB;
  const int gemm_blocks = NN / 16;  // 1250 blocks; 8 waves cover the 8 col tiles

  for (int t = 0; t < TT; ++t) {
    // p_l resets to p_0 at the start of every outer step
    copy_kernel<<<128, 256, 0, stream>>>(pbuf, p_0, NN);
    for (int l = 0; l < LL; ++l) {
      // th = h @ W_x[l]
      gemm_wmma_kernel<DD, false, false><<<gemm_blocks, 256, 0, stream>>>(
          hcur, nullptr, W_x + (size_t)l * DD * DD, th, NN);
      // agg = 0; agg[dst] += p[src] * relu(ins[batch[src]] * th[src])
      zero_kernel<<<2048, 256, 0, stream>>>(agg, NN * KK * DD);
      scatter_kernel<<<EE, 256, 0, stream>>>(edge_index, batch_idx, ins, th, pbuf, agg);
      // h_new = relu([h | agg] @ W_h[l])
      gemm_wmma_kernel<(KK + 1) * DD, true, true><<<gemm_blocks, 256, 0, stream>>>(
          hcur, agg, W_h + (size_t)l * (KK + 1) * DD * DD, hnxt, NN);
      // p = softmax(h_new @ w) over all nodes
      logits_kernel<<<(NN + 7) / 8, 256, 0, stream>>>(hnxt, w, logits, NN);
      softmax_reduce_kernel<<<1, 1024, 0, stream>>>(logits, stats, NN);
      softmax_norm_kernel<<<(NN + 255) / 256, 256, 0, stream>>>(logits, stats, pbuf, NN);
      float* tmp = hcur; hcur = hnxt; hnxt = tmp;
    }
    if (t < TT - 1) {  // final update_instructions does not affect p_out
      zero_kernel<<<16, 256, 0, stream>>>(he, BB * DD);
      he_accum_kernel<<<NN, DD, 0, stream>>>(hcur, p_0, batch_idx, he);
      ins_update_kernel<<<BB * KK, DD, 0, stream>>>(ins, he, W_q, gate);
    }
  }

  copy_kernel<<<128, 256, 0, stream>>>(out, pbuf, NN);
}